// Model_38963943309488
// MI455X (gfx1250) — compile-verified
//
#include <hip/hip_runtime.h>

typedef __attribute__((ext_vector_type(2))) float v2f;
typedef __attribute__((ext_vector_type(8))) float v8f;

#define N_NODES 50000
#define IN_CH   128
#define HID     128
#define OUT_CH  64
#define N_EDGES 800000

// ---------------------------------------------------------------------------
// Degree count + reciprocal
// ---------------------------------------------------------------------------
__global__ void count_edges(const int* __restrict__ dst, float* __restrict__ cnt) {
    int e = blockIdx.x * blockDim.x + threadIdx.x;
    if (e < N_EDGES) atomicAdd(&cnt[dst[e]], 1.0f);
}

__global__ void invert_cnt(float* __restrict__ cnt) {
    int i = blockIdx.x * blockDim.x + threadIdx.x;
    if (i < N_NODES) cnt[i] = 1.0f / fmaxf(cnt[i], 1.0f);
}

// ---------------------------------------------------------------------------
// Edge scatter: 32 lanes per edge, float4 per lane (128 channels)
// ---------------------------------------------------------------------------
__global__ void scatter_feats(const float* __restrict__ x,
                              const int* __restrict__ src,
                              const int* __restrict__ dst,
                              float* __restrict__ agg) {
    long long gid = (long long)blockIdx.x * blockDim.x + threadIdx.x;
    int e    = (int)(gid >> 5);
    int lane = (int)(gid & 31);
    if (e >= N_EDGES) return;
    int s = src[e];
    int d = dst[e];
    const float4 v = *(const float4*)(x + (long long)s * HID + lane * 4);
    float* o = agg + (long long)d * HID + lane * 4;
    atomicAdd(o + 0, v.x);
    atomicAdd(o + 1, v.y);
    atomicAdd(o + 2, v.z);
    atomicAdd(o + 3, v.w);
}

// ---------------------------------------------------------------------------
// Fused SAGE layer: out = (agg * inv_cnt) @ Wl^T + bl + xself @ Wr^T  [+ReLU]
// One wave computes one 16x16 output tile via V_WMMA_F32_16X16X4_F32.
// A 16x4 f32 fragment: lanes 0-15 hold K=k0+0/k0+1, lanes 16-31 K=k0+2/k0+3.
// B 4x16 f32 fragment (B[k][n] = W[n][k]): mirrored layout over lanes.
// C/D 16x16 f32: VGPR r = row r (lanes 0-15) / row r+8 (lanes 16-31).
// ---------------------------------------------------------------------------
__global__ void __launch_bounds__(128)
sage_gemm(const float* __restrict__ agg,      // [M,128] un-normalized sums
          const float* __restrict__ inv_cnt,  // [M]
          const float* __restrict__ xself,    // [M,128]
          const float* __restrict__ Wl,       // [128,128] row-major [n][k]
          const float* __restrict__ bl,       // [128]
          const float* __restrict__ Wr,       // [128,128]
          float* __restrict__ out,            // [M,128]
          int do_relu) {
    const int K = HID, N = HID;
    int wave = threadIdx.x >> 5;
    int lane = threadIdx.x & 31;
    int tile = blockIdx.x * 4 + wave;   // 3125*8 = 25000 tiles, grid 6250
    int m0 = (tile >> 3) * 16;          // N/16 == 8 tiles per row-band
    int n0 = (tile & 7) * 16;

    int hi   = lane >> 4;               // half-wave
    int l    = lane & 15;
    int arow = m0 + l;                  // per-lane A row (constant over K loop)
    int koff = hi * 2;                  // K sub-offset within the 4-wide step
    int wrow = n0 + l;                  // per-lane W row (= output column)

    float inv = inv_cnt[arow];

    const float* Aagg = agg   + (long long)arow * K + koff;
    const float* Aself= xself + (long long)arow * K + koff;
    const float* WlR  = Wl    + (long long)wrow * K + koff;
    const float* WrR  = Wr    + (long long)wrow * K + koff;

    v8f c = {0.f, 0.f, 0.f, 0.f, 0.f, 0.f, 0.f, 0.f};

    #pragma unroll 4
    for (int k0 = 0; k0 < K; k0 += 4) {
        v2f a, b;
        // neighbor-mean term
        a.x = Aagg[k0 + 0] * inv;
        a.y = Aagg[k0 + 1] * inv;
        b.x = WlR[k0 + 0];
        b.y = WlR[k0 + 1];
        c = __builtin_amdgcn_wmma_f32_16x16x4_f32(false, a, false, b,
                                                  (short)0, c, false, false);
        // self term
        a.x = Aself[k0 + 0];
        a.y = Aself[k0 + 1];
        b.x = WrR[k0 + 0];
        b.y = WrR[k0 + 1];
        c = __builtin_amdgcn_wmma_f32_16x16x4_f32(false, a, false, b,
                                                  (short)0, c, false, false);
    }

    float bias = bl[n0 + l];
    float* orow = out + (long long)(m0 + hi * 8) * N + n0 + l;
    #pragma unroll
    for (int r = 0; r < 8; ++r) {
        float v = c[r] + bias;
        if (do_relu) v = fmaxf(v, 0.0f);
        orow[(long long)r * N] = v;
    }
}

// ---------------------------------------------------------------------------
// Final linear head: out[M,64] = h @ Wout^T + bout (same WMMA tiling, N=64)
// ---------------------------------------------------------------------------
__global__ void __launch_bounds__(128)
out_gemm(const float* __restrict__ h,    // [M,128]
         const float* __restrict__ W,    // [64,128]
         const float* __restrict__ b,    // [64]
         float* __restrict__ out) {      // [M,64]
    const int K = HID, N = OUT_CH;
    int wave = threadIdx.x >> 5;
    int lane = threadIdx.x & 31;
    int tile = blockIdx.x * 4 + wave;   // 3125*4 tiles, grid 3125
    int m0 = (tile >> 2) * 16;
    int n0 = (tile & 3) * 16;

    int hi   = lane >> 4;
    int l    = lane & 15;
    int arow = m0 + l;
    int koff = hi * 2;
    int wrow = n0 + l;

    const float* Ar = h + (long long)arow * K + koff;
    const float* WR = W + (long long)wrow * K + koff;

    v8f c = {0.f, 0.f, 0.f, 0.f, 0.f, 0.f, 0.f, 0.f};

    #pragma unroll 4
    for (int k0 = 0; k0 < K; k0 += 4) {
        v2f a, bb;
        a.x  = Ar[k0 + 0];
        a.y  = Ar[k0 + 1];
        bb.x = WR[k0 + 0];
        bb.y = WR[k0 + 1];
        c = __builtin_amdgcn_wmma_f32_16x16x4_f32(false, a, false, bb,
                                                  (short)0, c, false, false);
    }

    float bias = b[n0 + l];
    float* orow = out + (long long)(m0 + hi * 8) * N + n0 + l;
    #pragma unroll
    for (int r = 0; r < 8; ++r) {
        orow[(long long)r * N] = c[r] + bias;
    }
}

// ---------------------------------------------------------------------------
// In-place log-softmax over rows of 64 (one wave32 per row, 2 cols/lane)
// ---------------------------------------------------------------------------
__global__ void log_softmax_rows(float* __restrict__ out) {
    int row  = blockIdx.x * (blockDim.x >> 5) + (threadIdx.x >> 5);
    int lane = threadIdx.x & 31;
    if (row >= N_NODES) return;
    float* p = out + (long long)row * OUT_CH;
    float v0 = p[lane];
    float v1 = p[lane + 32];
    float m = fmaxf(v0, v1);
    #pragma unroll
    for (int off = 16; off > 0; off >>= 1)
        m = fmaxf(m, __shfl_xor(m, off, 32));
    float s = expf(v0 - m) + expf(v1 - m);
    #pragma unroll
    for (int off = 16; off > 0; off >>= 1)
        s += __shfl_xor(s, off, 32);
    float lse = m + logf(s);
    p[lane]      = v0 - lse;
    p[lane + 32] = v1 - lse;
}

// ---------------------------------------------------------------------------
extern "C" void kernel_launch(void* const* d_in, const int* in_sizes, int n_in,
                              void* d_out, int out_size, void* d_ws, size_t ws_size,
                              hipStream_t stream) {
    const float* x    = (const float*)d_in[0];
    const int*   ei   = (const int*)d_in[1];   // [2, N_EDGES] int32
    const float* Wl1  = (const float*)d_in[2];
    const float* bl1  = (const float*)d_in[3];
    const float* Wr1  = (const float*)d_in[4];
    const float* Wl2  = (const float*)d_in[5];
    const float* bl2  = (const float*)d_in[6];
    const float* Wr2  = (const float*)d_in[7];
    const float* Wout = (const float*)d_in[8];
    const float* bout = (const float*)d_in[9];

    const int* src = ei;
    const int* dst = ei + N_EDGES;

    const size_t feat = (size_t)N_NODES * HID;
    float* agg = (float*)d_ws;          // 25.6 MB (reused for layer 2)
    float* h1  = agg + feat;            // 25.6 MB
    float* h2  = h1 + feat;             // 25.6 MB
    float* cnt = h2 + feat;             // 0.2 MB (holds 1/deg after invert)
    float* outp = (float*)d_out;        // [N_NODES, OUT_CH]

    const int SCATTER_BLOCKS = (N_EDGES * 32) / 256;        // 100000, exact
    const int SAGE_BLOCKS    = (N_NODES / 16) * (HID / 16) / 4;   // 6250
    const int OUT_BLOCKS     = (N_NODES / 16) * (OUT_CH / 16) / 4; // 3125

    // degree (shared by both layers)
    hipMemsetAsync(cnt, 0, (size_t)N_NODES * sizeof(float), stream);
    count_edges<<<(N_EDGES + 255) / 256, 256, 0, stream>>>(dst, cnt);
    invert_cnt<<<(N_NODES + 255) / 256, 256, 0, stream>>>(cnt);

    // layer 1
    hipMemsetAsync(agg, 0, feat * sizeof(float), stream);
    scatter_feats<<<SCATTER_BLOCKS, 256, 0, stream>>>(x, src, dst, agg);
    sage_gemm<<<SAGE_BLOCKS, 128, 0, stream>>>(agg, cnt, x, Wl1, bl1, Wr1, h1, 1);

    // layer 2
    hipMemsetAsync(agg, 0, feat * sizeof(float), stream);
    scatter_feats<<<SCATTER_BLOCKS, 256, 0, stream>>>(h1, src, dst, agg);
    sage_gemm<<<SAGE_BLOCKS, 128, 0, stream>>>(agg, cnt, h1, Wl2, bl2, Wr2, h2, 0);

    // head + log-softmax (in place on d_out)
    out_gemm<<<OUT_BLOCKS, 128, 0, stream>>>(h2, Wout, bout, outp);
    log_softmax_rows<<<(N_NODES + 7) / 8, 256, 0, stream>>>(outp);
}